// _GraspTargetLayer_54116587930265
// MI455X (gfx1250) — compile-verified
//
#include <hip/hip_runtime.h>
#include <math.h>

#define BS 32
#define KP 50000      // priors per batch
#define NG 32         // gt rows per batch
#define TILES (KP/16) // 3125 full 16-prior tiles per batch
#define XTH 16.0f     // FEAT_STRIDE/2
#define ANGTH 15.0f
#define EPSF 1e-14f

typedef float v2f __attribute__((ext_vector_type(2)));
typedef float v8f __attribute__((ext_vector_type(8)));

__device__ __forceinline__ unsigned keyxf(float f) {
  // monotone float -> uint mapping (total order matching float <)
  unsigned u = __float_as_uint(f);
  return (u & 0x80000000u) ? ~u : (u | 0x80000000u);
}

// ------------------------------------------------------------------
// K1: anchor matching + WMMA scatter-reduce + grasp encode + loss_c
// one wave per 16-prior tile; 8 waves / block; grid = (391, 32)
// D = sum_s A_s(16x4 match flags) x B_s(4x16 [gt cols | ones]) + EPS
// ------------------------------------------------------------------
__global__ __launch_bounds__(256) void k1_match_encode(
    const float* __restrict__ conf, const float* __restrict__ gt,
    const float* __restrict__ priors, float* __restrict__ out_loc,
    float* __restrict__ ws_loss)
{
  __shared__ float4 s_m4[NG];      // {gx, gy, gangle, valid_mask}
  __shared__ float  s_b[NG * 16];  // padded B: cols 0..4 = gt, 5 = 1.0, 6..15 = 0
  __shared__ float  s_d[8 * 256];  // per-wave 8 VGPRs x 32 lanes of D

  const int tid = threadIdx.x;
  const int b   = blockIdx.y;

  if (tid < NG) {
    const float* g = gt + ((size_t)b * NG + tid) * 5;
    float g0 = g[0], g1 = g[1], g2 = g[2], g3 = g[3], g4 = g[4];
    int allz = (g0 == 0.f) & (g1 == 0.f) & (g2 == 0.f) & (g3 == 0.f) & (g4 == 0.f);
    s_m4[tid] = make_float4(g0, g1, g4, allz ? 0.f : 1.f);
  }
  // fill padded 32x16 B matrix (one-off setup, 2 strided iterations)
  for (int i = tid; i < NG * 16; i += 256) {
    int row = i >> 4, col = i & 15;
    float v = 0.0f;
    if (col < 5)       v = gt[((size_t)b * NG + row) * 5 + col];
    else if (col == 5) v = 1.0f;
    s_b[i] = v;
  }
  __syncthreads();

  const int wave = tid >> 5;
  const int lane = tid & 31;
  const int tile = blockIdx.x * 8 + wave;
  if (tile >= TILES) return;          // wave-uniform exit

  const int  n    = lane & 15;        // B/C/D column for this lane
  const bool hi   = lane >= 16;
  const int  kofs = hi ? 2 : 0;       // K-pair offset for this half-wave
  const int  m    = tile * 16 + n;    // prior row this lane owns

  const float* pr = priors + ((size_t)b * KP + m) * 5;
  const float px = pr[0], py = pr[1], pw = pr[2], ph = pr[3], pa = pr[4];

  // cover the tail conf load with the matrix work
  __builtin_prefetch(conf + ((size_t)b * KP + m) * 2, 0, 0);

  // C init: columns 0..4 get +EPS (loc = einsum + EPS), col 5 counts matches
  v8f c;
  {
    const float ci = (n < 5) ? EPSF : 0.0f;
    #pragma unroll
    for (int j = 0; j < 8; j++) c[j] = ci;
  }

  #pragma unroll
  for (int s = 0; s < 8; s++) {
    const int k0 = 4 * s + kofs;
    // branchless match flags (no EXEC churn around WMMA)
    const float4 gA = s_m4[k0];
    const float4 gB = s_m4[k0 + 1];
    const int ok0 = (fabsf(px - gA.x) <= XTH) & (fabsf(py - gA.y) <= XTH) &
                    (fabsf(pa - gA.z) <= ANGTH);
    const int ok1 = (fabsf(px - gB.x) <= XTH) & (fabsf(py - gB.y) <= XTH) &
                    (fabsf(pa - gB.z) <= ANGTH);
    v2f A;  A[0]  = ok0 ? gA.w : 0.0f;
            A[1]  = ok1 ? gB.w : 0.0f;
    v2f Bv; Bv[0] = s_b[k0 * 16 + n];          // unconditional padded loads
            Bv[1] = s_b[(k0 + 1) * 16 + n];
    c = __builtin_amdgcn_wmma_f32_16x16x4_f32(
        false, A, false, Bv, (short)0, c, false, false);
  }

  // transpose D (rows striped over lanes) back to one-row-per-lane via LDS
  {
    float* sd = s_d + wave * 256;
    #pragma unroll
    for (int j = 0; j < 8; j++) sd[j * 32 + lane] = c[j];
  }
  asm volatile("s_wait_dscnt 0" ::: "memory");

  if (!hi) {
    const float* sd = s_d + wave * 256;
    const int base = (n < 8) ? (n * 32) : ((n - 8) * 32 + 16);
    float v0 = sd[base + 0], v1 = sd[base + 1], v2 = sd[base + 2];
    float v3 = sd[base + 3], v4 = sd[base + 4], v5 = sd[base + 5];
    const bool  pos = v5 > 0.5f;                 // match_num > 0
    const float den = fmaxf(v5, 1.0f);
    const float l0 = pos ? v0 / den : v0;
    const float l1 = pos ? v1 / den : v1;
    const float l2 = pos ? v2 / den : v2;
    const float l3 = pos ? v3 / den : v3;
    const float l4 = pos ? v4 / den : v4;
    // grasp_encode + (x - MEAN)/STD, MEANS = 0, STDS = .1 .1 .2 .2 .1
    float o0 = ((l0 - px) / pw) * 10.0f;
    float o1 = ((l1 - py) / ph) * 10.0f;
    float o2 = logf(l2 / pw) * 5.0f;
    float o3 = logf(l3 / ph) * 5.0f;
    float o4 = (l4 - pa) * 10.0f;
    float* o = out_loc + ((size_t)b * KP + m) * 5;
    o[0] = o0; o[1] = o1; o[2] = o2; o[3] = o3; o[4] = o4;
    // hard-negative loss: lse - c[label]; pos entries get exact -1 sentinel
    const float* cf = conf + ((size_t)b * KP + m) * 2;
    float c0 = cf[0], c1 = cf[1];
    float mx  = fmaxf(c0, c1);
    float lse = mx + logf(expf(c0 - mx) + expf(c1 - mx));
    float loss = pos ? -1.0f : (lse - c0);       // label=0 for non-pos
    ws_loss[(size_t)b * KP + m] = loss;
  }
}

// ------------------------------------------------------------------
// K2: per-batch exact radix-select of the num_neg-th largest loss
// grid = 32 (1 block / batch), 1024 threads
// meta[b] = { pivot_key, take_eq, num_pos, pad }
// ------------------------------------------------------------------
__global__ __launch_bounds__(1024) void k2_select(
    const float* __restrict__ ws_loss, unsigned* __restrict__ ws_meta)
{
  __shared__ unsigned hist[2048];
  __shared__ unsigned sh_np;
  __shared__ unsigned sh_chosen, sh_t;

  const int tid = threadIdx.x;
  const int b   = blockIdx.x;
  const float* loss = ws_loss + (size_t)b * KP;

  if (tid == 0) sh_np = 0;
  __syncthreads();
  unsigned local = 0;
  for (int i = tid; i < KP; i += 1024)
    if (loss[i] == -1.0f) local++;
  if (local) atomicAdd(&sh_np, local);
  __syncthreads();

  const unsigned np = sh_np;
  unsigned num_neg = 3u * np;
  if (num_neg > (unsigned)KP) num_neg = KP;
  if (num_neg == 0) {
    if (tid == 0) {
      ws_meta[4 * b + 0] = 0xFFFFFFFFu;  // nothing exceeds max key
      ws_meta[4 * b + 1] = 0u;           // take_eq
      ws_meta[4 * b + 2] = np;
    }
    return;
  }

  const int shifts[3] = {21, 10, 0};
  const int bitsl[3]  = {11, 11, 10};
  unsigned t = num_neg, prefix = 0;
  int prefbits = 0;

  for (int lvl = 0; lvl < 3; lvl++) {
    const int nb = 1 << bitsl[lvl];
    for (int i = tid; i < 2048; i += 1024) hist[i] = 0;
    __syncthreads();
    for (int i = tid; i < KP; i += 1024) {
      unsigned key = keyxf(loss[i]);
      if (prefbits == 0 || (key >> (32 - prefbits)) == prefix) {
        unsigned bin = (key >> shifts[lvl]) & (unsigned)(nb - 1);
        atomicAdd(&hist[bin], 1u);
      }
    }
    __syncthreads();
    if (tid == 0) {
      unsigned cum = 0; int chosen = 0;
      for (int bin = nb - 1; bin >= 0; --bin) {
        unsigned cnt = hist[bin];
        if (cum + cnt >= t) { chosen = bin; break; }
        cum += cnt;
      }
      sh_chosen = (unsigned)chosen;
      sh_t = t - cum;
    }
    __syncthreads();
    prefix = (prefix << bitsl[lvl]) | sh_chosen;
    t = sh_t;
    prefbits += bitsl[lvl];
    __syncthreads();
  }

  if (tid == 0) {
    ws_meta[4 * b + 0] = prefix;  // exact 32-bit pivot key
    ws_meta[4 * b + 1] = t;       // how many pivot-equal elems to take (index order)
    ws_meta[4 * b + 2] = np;
  }
}

// ------------------------------------------------------------------
// K3a: bulk per-prior outputs (conf_t / iw / ow); neg = key > pivot
// grid = (196, 32), 256 threads
// ------------------------------------------------------------------
__global__ __launch_bounds__(256) void k3a_outputs(
    const float* __restrict__ ws_loss, const unsigned* __restrict__ ws_meta,
    float* __restrict__ out_conf, float* __restrict__ out_iw,
    float* __restrict__ out_ow)
{
  const int b = blockIdx.y;
  const int k = blockIdx.x * 256 + threadIdx.x;
  if (k >= KP) return;

  const unsigned pivot = ws_meta[4 * b + 0];
  const unsigned np    = ws_meta[4 * b + 2];
  const float loss = ws_loss[(size_t)b * KP + k];
  const bool pos = (loss == -1.0f);
  const bool neg = !pos && (keyxf(loss) > pivot);

  out_conf[(size_t)b * KP + k] = pos ? 1.0f : (neg ? 0.0f : -1.0f);

  const float iwv = pos ? 1.0f : 0.0f;                 // BBOX_INSIDE_WEIGHT=1
  const float den = (float)((4u * np) | 1u);           // (num_pos+3*num_pos)|1
  const float owv = (pos || neg) ? (1.0f / den) : 0.0f;
  float* iw = out_iw + ((size_t)b * KP + k) * 5;
  float* ow = out_ow + ((size_t)b * KP + k) * 5;
  #pragma unroll
  for (int d = 0; d < 5; d++) { iw[d] = iwv; ow[d] = owv; }
}

// ------------------------------------------------------------------
// K3b: stable tie-break for loss == pivot: take first take_eq in index
// order (matches jnp stable double-argsort). grid = 32, 1024 threads.
// ------------------------------------------------------------------
__global__ __launch_bounds__(1024) void k3b_fixup(
    const float* __restrict__ ws_loss, const unsigned* __restrict__ ws_meta,
    float* __restrict__ out_conf, float* __restrict__ out_ow)
{
  const int tid = threadIdx.x;
  const int b   = blockIdx.x;
  const unsigned pivot = ws_meta[4 * b + 0];
  const unsigned take  = ws_meta[4 * b + 1];
  const unsigned np    = ws_meta[4 * b + 2];
  if (take == 0) return;
  const float inv = 1.0f / (float)((4u * np) | 1u);

  __shared__ unsigned wtot[32];
  __shared__ unsigned runbase;
  if (tid == 0) runbase = 0;
  __syncthreads();

  const int lane = tid & 31, wid = tid >> 5;
  for (int base = 0; base < KP; base += 1024) {
    const int idx = base + tid;
    bool eq = false, pos = false;
    if (idx < KP) {
      float l = ws_loss[(size_t)b * KP + idx];
      pos = (l == -1.0f);
      eq  = (keyxf(l) == pivot);
    }
    unsigned mask  = (unsigned)__ballot(eq);           // wave32: low 32 bits
    unsigned below = __popc(mask & ((1u << lane) - 1u));
    if (lane == 0) wtot[wid] = __popc(mask);
    __syncthreads();
    if (tid == 0) {
      unsigned s = runbase;
      for (int w = 0; w < 32; w++) { unsigned tv = wtot[w]; wtot[w] = s; s += tv; }
      runbase = s;
    }
    __syncthreads();
    unsigned rank = wtot[wid] + below;
    if (eq && !pos && rank < take) {
      out_conf[(size_t)b * KP + idx] = 0.0f;           // becomes neg
      float* ow = out_ow + ((size_t)b * KP + idx) * 5;
      #pragma unroll
      for (int d = 0; d < 5; d++) ow[d] = inv;
    }
    __syncthreads();
  }
}

// ------------------------------------------------------------------
extern "C" void kernel_launch(void* const* d_in, const int* in_sizes, int n_in,
                              void* d_out, int out_size, void* d_ws, size_t ws_size,
                              hipStream_t stream) {
  (void)in_sizes; (void)n_in; (void)out_size; (void)ws_size;
  const float* conf   = (const float*)d_in[0];  // [32,50000,2]
  const float* gt     = (const float*)d_in[1];  // [32,32,5]
  const float* priors = (const float*)d_in[2];  // [32,50000,5]

  float* out      = (float*)d_out;
  float* out_loc  = out;                                    // 8,000,000
  float* out_conf = out + (size_t)BS * KP * 5;              // 1,600,000
  float* out_iw   = out_conf + (size_t)BS * KP;             // 8,000,000
  float* out_ow   = out_iw + (size_t)BS * KP * 5;           // 8,000,000

  float*    ws_loss = (float*)d_ws;                         // 6.4 MB
  unsigned* ws_meta = (unsigned*)((char*)d_ws + (size_t)BS * KP * sizeof(float));

  dim3 g1((TILES + 7) / 8, BS);
  k1_match_encode<<<g1, 256, 0, stream>>>(conf, gt, priors, out_loc, ws_loss);

  k2_select<<<BS, 1024, 0, stream>>>(ws_loss, ws_meta);

  dim3 g3((KP + 255) / 256, BS);
  k3a_outputs<<<g3, 256, 0, stream>>>(ws_loss, ws_meta, out_conf, out_iw, out_ow);

  k3b_fixup<<<BS, 1024, 0, stream>>>(ws_loss, ws_meta, out_conf, out_ow);
}